// RGCN_78219944394964
// MI455X (gfx1250) — compile-verified
//
#include <hip/hip_runtime.h>

#define N_NODES   100000
#define N_REL     8
#define HID       16
#define N_EDGES   3200000

typedef __attribute__((ext_vector_type(2))) float v2f;
typedef __attribute__((ext_vector_type(8))) float v8f;

// ---------------------------------------------------------------- utilities
__global__ void zero_cnt_kernel(int* __restrict__ cnt) {
    int i = blockIdx.x * 256 + threadIdx.x;
    if (i < N_NODES * N_REL) cnt[i] = 0;
}

__global__ void count_edges_kernel(const int* __restrict__ dst,
                                   const int* __restrict__ et,
                                   int* __restrict__ cnt) {
    int e = blockIdx.x * 256 + threadIdx.x;
    if (e < N_EDGES) atomicAdd(&cnt[dst[e] * N_REL + et[e]], 1);
}

__global__ void make_recip_kernel(const int* __restrict__ cnt,
                                  float* __restrict__ recip) {
    int i = blockIdx.x * 256 + threadIdx.x;
    if (i < N_NODES * N_REL) {
        int c = cnt[i];
        recip[i] = 1.0f / (float)(c > 0 ? c : 1);
    }
}

// ------------------------------------------------- per-tile WMMA transform
// One wave (32 lanes) owns a 16-node tile. Computes xW[r] = x@W[r] for the 8
// relations, plus agg_init = x@root + bias, via V_WMMA_F32_16X16X4_F32
// (K=16 done as 4 chained K=4 WMMAs). W/root staged transposed in LDS.
template <bool RELU>
__global__ void rgcn_transform_kernel(const float* __restrict__ x,     // [N,16]
                                      const float* __restrict__ W,     // [8,16,16]
                                      const float* __restrict__ root,  // [16,16]
                                      const float* __restrict__ bias,  // [16]
                                      float* __restrict__ xW,          // [8,N,16]
                                      float* __restrict__ agg)         // [N,16]
{
    __shared__ float ldsW[9 * 256];   // transposed: ldsW[mat][n][k]
    __shared__ float ldsB[16];

    const int tid = threadIdx.x;
    for (int i = tid; i < 9 * 256; i += 256) {
        int mat = i >> 8;
        int idx = i & 255;
        int k = idx >> 4, n = idx & 15;
        float v = (mat < 8) ? W[i] : root[idx];
        ldsW[mat * 256 + n * 16 + k] = v;   // transpose k<->n
    }
    if (tid < 16) ldsB[tid] = bias[tid];
    __syncthreads();

    const int wave = tid >> 5;
    const int lane = tid & 31;
    const int tile = blockIdx.x * 8 + wave;
    const int ntiles = N_NODES / 16;          // 6250, exact
    if (tile >= ntiles) return;               // wave-uniform: EXEC stays full

    const int row0  = tile * 16;
    const int m     = lane & 15;
    const int khalf = (lane >> 4) << 1;       // A/B: lanes 0-15 -> K{0,1}, 16-31 -> K{2,3}
    const int rhalf = (lane >> 4) << 3;       // D:   lanes 0-15 -> M=v,    16-31 -> M=v+8

    // A fragments for all four K=4 slabs of this 16x16 input tile
    const float* xrow = x + (size_t)(row0 + m) * HID;
    v2f a[4];
#pragma unroll
    for (int ki = 0; ki < 4; ++ki) {
        float ax = xrow[4 * ki + khalf];
        float ay = xrow[4 * ki + khalf + 1];
        if (RELU) { ax = fmaxf(ax, 0.0f); ay = fmaxf(ay, 0.0f); }
        a[ki].x = ax; a[ki].y = ay;
    }

    const int n = m;                          // output column owned by this lane
#pragma unroll
    for (int mat = 0; mat < 9; ++mat) {
        const float* wt = &ldsW[mat * 256 + n * 16];
        v8f c = {0.f, 0.f, 0.f, 0.f, 0.f, 0.f, 0.f, 0.f};
#pragma unroll
        for (int ki = 0; ki < 4; ++ki) {
            v2f b;
            b.x = wt[4 * ki + khalf];         // contiguous -> ds_load_b64
            b.y = wt[4 * ki + khalf + 1];
            c = __builtin_amdgcn_wmma_f32_16x16x4_f32(
                    false, a[ki], false, b, (short)0, c, false, false);
        }
        if (mat < 8) {
            float* op = xW + (size_t)mat * N_NODES * HID;
#pragma unroll
            for (int v = 0; v < 8; ++v)
                op[(size_t)(row0 + rhalf + v) * HID + n] = c[v];
        } else {
            float bn = ldsB[n];
#pragma unroll
            for (int v = 0; v < 8; ++v)
                agg[(size_t)(row0 + rhalf + v) * HID + n] = c[v] + bn;
        }
    }
}

// ------------------------------------------------------- edge gather/scatter
// 16 lanes per edge: coalesced 64B row gather, consecutive-address f32 atomics.
__global__ void edge_scatter_kernel(const int* __restrict__ src,
                                    const int* __restrict__ dst,
                                    const int* __restrict__ et,
                                    const float* __restrict__ recip,
                                    const float* __restrict__ xW,   // [8,N,16]
                                    float* __restrict__ agg)        // [N,16]
{
    long long gid = (long long)blockIdx.x * 256 + threadIdx.x;
    if (gid >= (long long)N_EDGES * HID) return;
    int e = (int)(gid >> 4);
    int j = (int)(gid & 15);
    int t = et[e];
    int s = src[e];
    int d = dst[e];
    float w   = recip[d * N_REL + t];
    float msg = xW[((size_t)t * N_NODES + s) * HID + j];
    atomicAdd(&agg[(size_t)d * HID + j], msg * w);
}

// ------------------------------------------------------------- log_softmax
// 16 lanes per node row; xor masks < 16 stay inside the aligned 16-lane group.
__global__ void logsoftmax16_kernel(const float* __restrict__ h,
                                    float* __restrict__ out)
{
    int gid = blockIdx.x * 256 + threadIdx.x;
    if (gid >= N_NODES * HID) return;
    float v  = h[gid];
    float mx = v;
#pragma unroll
    for (int s = 1; s < 16; s <<= 1)
        mx = fmaxf(mx, __shfl_xor(mx, s, 32));
    float ex  = __expf(v - mx);
    float sum = ex;
#pragma unroll
    for (int s = 1; s < 16; s <<= 1)
        sum += __shfl_xor(sum, s, 32);
    out[gid] = v - mx - __logf(sum);
}

// ------------------------------------------------------------------ launch
extern "C" void kernel_launch(void* const* d_in, const int* in_sizes, int n_in,
                              void* d_out, int out_size, void* d_ws, size_t ws_size,
                              hipStream_t stream)
{
    const float* embed = (const float*)d_in[0];
    const float* W1    = (const float*)d_in[1];
    const float* root1 = (const float*)d_in[2];
    const float* b1    = (const float*)d_in[3];
    const float* W2    = (const float*)d_in[4];
    const float* root2 = (const float*)d_in[5];
    const float* b2    = (const float*)d_in[6];
    const int*   eidx  = (const int*)d_in[7];   // [2, E]
    const int*   etype = (const int*)d_in[8];
    const int*   src   = eidx;
    const int*   dst   = eidx + N_EDGES;
    float*       out   = (float*)d_out;

    // workspace carve-up (~70.4 MB total; L2 is 192 MB)
    char* ws = (char*)d_ws;
    int*   cnt   = (int*)ws;    ws += (size_t)N_NODES * N_REL * sizeof(int);
    float* recip = (float*)ws;  ws += (size_t)N_NODES * N_REL * sizeof(float);
    float* xW    = (float*)ws;  ws += (size_t)N_REL * N_NODES * HID * sizeof(float);
    float* agg1  = (float*)ws;  ws += (size_t)N_NODES * HID * sizeof(float);
    float* agg2  = (float*)ws;

    const int nr_blocks   = (N_NODES * N_REL) / 256;          // 3125 exact
    const int cnt_blocks  = (N_EDGES + 255) / 256;            // 12500
    const int tile_blocks = (N_NODES / 16 + 7) / 8;           // 782 (8 waves/block)
    const int edge_blocks = (int)(((long long)N_EDGES * HID) / 256); // 200000 exact
    const int sm_blocks   = (N_NODES * HID) / 256;            // 6250 exact

    // edge-structure normalization (shared by both layers)
    zero_cnt_kernel   <<<nr_blocks,  256, 0, stream>>>(cnt);
    count_edges_kernel<<<cnt_blocks, 256, 0, stream>>>(dst, etype, cnt);
    make_recip_kernel <<<nr_blocks,  256, 0, stream>>>(cnt, recip);

    // layer 1
    rgcn_transform_kernel<false><<<tile_blocks, 256, 0, stream>>>(
        embed, W1, root1, b1, xW, agg1);
    edge_scatter_kernel<<<edge_blocks, 256, 0, stream>>>(
        src, dst, etype, recip, xW, agg1);

    // layer 2 (ReLU fused into the A-fragment load)
    rgcn_transform_kernel<true><<<tile_blocks, 256, 0, stream>>>(
        agg1, W2, root2, b2, xW, agg2);
    edge_scatter_kernel<<<edge_blocks, 256, 0, stream>>>(
        src, dst, etype, recip, xW, agg2);

    // log_softmax
    logsoftmax16_kernel<<<sm_blocks, 256, 0, stream>>>(agg2, out);
}